// GraphConvolution_36893769073012
// MI455X (gfx1250) — compile-verified
//
#include <hip/hip_runtime.h>
#include <hip/hip_bf16.h>

// ---------------------------------------------------------------------------
// GraphConvolution forward for MI455X (gfx1250, wave32, WMMA).
//   agg = segment_sum(edge_w * x[col], row)      (atomic scatter, f32, L2-resident)
//   out = [agg | x] @ W + b                      (V_WMMA_F32_16X16X4_F32, LDS-staged)
//   out = relu(batchnorm(out))                   (BN stats fused into GEMM epilogue)
// agg lives inside d_out (same N x 128 f32 shape); ws holds only ~2 KB of stats.
// ---------------------------------------------------------------------------

typedef float v2f __attribute__((ext_vector_type(2)));
typedef float v8f __attribute__((ext_vector_type(8)));
typedef int   v4i __attribute__((vector_size(16)));                 // matches builtin proto
typedef __attribute__((address_space(1))) v4i* as1_v4i;             // global
typedef __attribute__((address_space(3))) v4i* as3_v4i;             // LDS

#define DFEAT   128
#define KDIM    256
#define KCHUNK  64
#define ASTR    260        // A-tile LDS row stride (floats); 260 % 64 = 4 -> bank fan-out
#define BN_EPS  1e-5f

// --- CDNA5 async global->LDS path (builtin confirmed present; proto from
//     hipcc diagnostic: (v4i AS1*, v4i AS3*, Ii offset, Ii cpol)) ------------
#if defined(__AMDGCN__) && __has_builtin(__builtin_amdgcn_global_load_async_to_lds_b128)
#define HAVE_ASYNC_LDS 1
#else
#define HAVE_ASYNC_LDS 0
#endif

__device__ __forceinline__ void stage16(const void* gsrc, void* ldst) {
#if HAVE_ASYNC_LDS
    __builtin_amdgcn_global_load_async_to_lds_b128(
        (as1_v4i)(v4i*)gsrc, (as3_v4i)(v4i*)ldst, 0, 0);
#else
    *(float4*)ldst = *(const float4*)gsrc;
#endif
}

__device__ __forceinline__ void stage_wait() {
#if HAVE_ASYNC_LDS && __has_builtin(__builtin_amdgcn_s_wait_asynccnt)
    __builtin_amdgcn_s_wait_asynccnt(0);
#endif
}

// ---------------------------------------------------------------------------
// Kernel 1: zero agg region (d_out) and BN accumulators (ws).
// ---------------------------------------------------------------------------
__global__ __launch_bounds__(256) void gc_zero(float4* __restrict__ out4, int n4,
                                               float* __restrict__ stats, int nstats) {
    int i = blockIdx.x * 256 + threadIdx.x;
    if (i < n4) out4[i] = make_float4(0.f, 0.f, 0.f, 0.f);
    if (i < nstats) stats[i] = 0.f;
}

// ---------------------------------------------------------------------------
// Kernel 2: SpMM scatter. One wave per edge, 4 edges per wave with prefetch
// of the next source row (x is 51 MB -> L2-resident; prefetch hides L2 hit
// latency). Each lane: float4 gather + 4 global f32 atomics.
// ---------------------------------------------------------------------------
#define EDGES_PER_WAVE 4

__global__ __launch_bounds__(256) void gc_spmm(const float* __restrict__ x,
                                               const float* __restrict__ ew,
                                               const int*   __restrict__ erow,
                                               const int*   __restrict__ ecol,
                                               float*       __restrict__ agg,
                                               int n_edges) {
    const int lane = threadIdx.x & 31;
    const int gw   = blockIdx.x * 8 + (threadIdx.x >> 5);
    const int e0   = gw * EDGES_PER_WAVE;

    #pragma unroll
    for (int j = 0; j < EDGES_PER_WAVE; ++j) {
        const int e = e0 + j;
        if (e >= n_edges) break;
        // prefetch next edge's source row while we crunch this one
        if (j + 1 < EDGES_PER_WAVE && e + 1 < n_edges) {
            const int cn = ecol[e + 1];
            __builtin_prefetch(x + (long long)cn * DFEAT + lane * 4, 0, 3);
        }
        const int   r = erow[e];
        const int   c = ecol[e];
        const float w = ew[e];
        const float4 xv = ((const float4*)(x + (long long)c * DFEAT))[lane];
        float* dst = agg + (long long)r * DFEAT + lane * 4;
        atomicAdd(dst + 0, xv.x * w);
        atomicAdd(dst + 1, xv.y * w);
        atomicAdd(dst + 2, xv.z * w);
        atomicAdd(dst + 3, xv.w * w);
    }
}

// ---------------------------------------------------------------------------
// Kernel 3: WMMA fp32 GEMM + bias, BN partial sums fused.
// Block = 256 threads = 8 waves; computes a 16-row x 128-col output strip.
// Stage 1: full 16x256 support tile ([agg|x]) -> LDS via async-to-LDS
//          (shared by all 8 waves; agg/x select hoisted out of the K loop).
// Stage 2: W in four 64x128 K-chunks -> LDS (async).
// Wave w owns the 16x16 tile at columns [16w,16w+16); 64 WMMAs over K=256.
// ---------------------------------------------------------------------------
__global__ __launch_bounds__(256) void gc_gemm_wmma(const float* __restrict__ x,
                                                    const float* __restrict__ W,
                                                    const float* __restrict__ bias,
                                                    float*       __restrict__ out,   // in: agg; out: linear pre-BN
                                                    float*       __restrict__ bn_sum,
                                                    float*       __restrict__ bn_sumsq) {
    __shared__ float Alds[16 * ASTR];       // 16.25 KB support tile (K = 0..255)
    __shared__ float Wlds[KCHUNK * DFEAT];  // 32 KB   W K-chunk

    const int tid  = threadIdx.x;
    const int wave = tid >> 5;
    const int lane = tid & 31;
    const int lo   = lane & 15;   // N index within tile / A row select
    const int hi   = lane >> 4;   // K sub-select for A/B fragments
    const int m0   = blockIdx.x * 16;
    const int n0   = wave * 16;

    // ---- stage support tile: rows m0..m0+15, cols 0..255 = [agg | x] ------
    // 16 rows x 64 float4 = 1024 stages; 4 per thread, coalesced per row.
    for (int i = tid; i < 16 * 64; i += 256) {
        const int r = i >> 6;        // row 0..15
        const int q = i & 63;        // float4 column quad (k = 4q)
        const long long row = (long long)(m0 + r) * DFEAT;
        const float4* src = (q < 32) ? ((const float4*)(out + row) + q)
                                     : ((const float4*)(x + row) + (q - 32));
        stage16(src, Alds + r * ASTR + q * 4);
    }
    stage_wait();
    __syncthreads();   // support tile snapshotted; `out` rows now writable

    v8f acc = {0.f, 0.f, 0.f, 0.f, 0.f, 0.f, 0.f, 0.f};

    const int aoff = lo * ASTR + 2 * hi;           // A fragment base (float idx)
    const int boff = (2 * hi) * DFEAT + n0 + lo;   // B fragment base (float idx)

    for (int kc = 0; kc < KDIM; kc += KCHUNK) {
        __syncthreads();   // protect Wlds reuse across chunks
        for (int i = tid; i < (KCHUNK * DFEAT) / 4; i += 256) {
            stage16((const float4*)(W + kc * DFEAT) + i, Wlds + i * 4);
        }
        stage_wait();
        __syncthreads();

        #pragma unroll
        for (int kk = 0; kk < KCHUNK; kk += 4) {
            v2f a;   // support[m0+lo][kc+kk+2hi .. +1]
            a.x = Alds[aoff + kc + kk];
            a.y = Alds[aoff + kc + kk + 1];
            v2f b;   // W[kc+kk+2hi .. +1][n0+lo]
            b.x = Wlds[boff + kk * DFEAT];
            b.y = Wlds[boff + kk * DFEAT + DFEAT];
            acc = __builtin_amdgcn_wmma_f32_16x16x4_f32(
                /*neg_a=*/false, a, /*neg_b=*/false, b,
                /*c_mod=*/(short)0, acc, /*reuse_a=*/false, /*reuse_b=*/false);
        }
    }

    // ---- epilogue: bias, store pre-BN, fused BN partial stats -------------
    const float bc = bias[n0 + lo];
    float s = 0.f, q = 0.f;
    #pragma unroll
    for (int v = 0; v < 8; ++v) {
        float val = acc[v] + bc;
        int m = m0 + v + 8 * hi;                 // C/D layout: VGPR v, lane-half hi
        out[(long long)m * DFEAT + n0 + lo] = val;
        s += val;
        q += val * val;
    }
    atomicAdd(&bn_sum[n0 + lo], s);
    atomicAdd(&bn_sumsq[n0 + lo], q);
}

// ---------------------------------------------------------------------------
// Kernel 4: fold BN stats into per-channel scale/shift.
// ---------------------------------------------------------------------------
__global__ void gc_finalize(const float* __restrict__ bn_sum,
                            const float* __restrict__ bn_sumsq,
                            const float* __restrict__ gamma,
                            const float* __restrict__ beta,
                            float* __restrict__ scale,
                            float* __restrict__ shift,
                            float inv_n) {
    int c = threadIdx.x;  // 128 threads
    float mean = bn_sum[c] * inv_n;
    float var  = bn_sumsq[c] * inv_n - mean * mean;
    float inv  = rsqrtf(var + BN_EPS);
    float sc   = gamma[c] * inv;
    scale[c] = sc;
    shift[c] = beta[c] - mean * sc;
}

// ---------------------------------------------------------------------------
// Kernel 5: in-place BN apply + ReLU, float4 vectorized.
// ---------------------------------------------------------------------------
__global__ __launch_bounds__(256) void gc_bn_relu(float4* __restrict__ out4,
                                                  const float* __restrict__ scale,
                                                  const float* __restrict__ shift,
                                                  int n4) {
    int i = blockIdx.x * 256 + threadIdx.x;
    if (i >= n4) return;
    int c = (i * 4) & (DFEAT - 1);
    float4 v = out4[i];
    v.x = fmaxf(fmaf(v.x, scale[c + 0], shift[c + 0]), 0.f);
    v.y = fmaxf(fmaf(v.y, scale[c + 1], shift[c + 1]), 0.f);
    v.z = fmaxf(fmaf(v.z, scale[c + 2], shift[c + 2]), 0.f);
    v.w = fmaxf(fmaf(v.w, scale[c + 3], shift[c + 3]), 0.f);
    out4[i] = v;
}

// ---------------------------------------------------------------------------
// Host entry
// ---------------------------------------------------------------------------
extern "C" void kernel_launch(void* const* d_in, const int* in_sizes, int n_in,
                              void* d_out, int out_size, void* d_ws, size_t ws_size,
                              hipStream_t stream) {
    const float* x     = (const float*)d_in[0];   // [N,128]
    const float* ew    = (const float*)d_in[1];   // [E]
    const float* W     = (const float*)d_in[2];   // [256,128]
    const float* bias  = (const float*)d_in[3];   // [128]
    const float* gamma = (const float*)d_in[4];   // [128]
    const float* beta  = (const float*)d_in[5];   // [128]
    const int*   erow  = (const int*)d_in[6];     // [E]
    const int*   ecol  = (const int*)d_in[7];     // [E]

    const int n_nodes = in_sizes[0] / DFEAT;
    const int n_edges = in_sizes[1];

    float* out = (float*)d_out;                   // doubles as agg buffer
    float* stats    = (float*)d_ws;               // 512 floats
    float* bn_sum   = stats;
    float* bn_sumsq = stats + DFEAT;
    float* scale    = stats + 2 * DFEAT;
    float* shift    = stats + 3 * DFEAT;

    const int n4 = n_nodes * DFEAT / 4;

    // 1) zero agg (in d_out) + BN accumulators
    gc_zero<<<(n4 + 255) / 256, 256, 0, stream>>>((float4*)out, n4, bn_sum, 2 * DFEAT);

    // 2) SpMM scatter: 8 waves/block, 4 edges/wave with prefetch
    const int waves_needed = (n_edges + EDGES_PER_WAVE - 1) / EDGES_PER_WAVE;
    gc_spmm<<<(waves_needed + 7) / 8, 256, 0, stream>>>(x, ew, erow, ecol, out, n_edges);

    // 3) WMMA GEMM + bias + fused BN partial stats (16-row strips)
    gc_gemm_wmma<<<n_nodes / 16, 256, 0, stream>>>(x, W, bias, out, bn_sum, bn_sumsq);

    // 4) finalize BN scale/shift
    gc_finalize<<<1, DFEAT, 0, stream>>>(bn_sum, bn_sumsq, gamma, beta, scale, shift,
                                         1.0f / (float)n_nodes);

    // 5) BN apply + ReLU in place
    gc_bn_relu<<<(n4 + 255) / 256, 256, 0, stream>>>((float4*)out, scale, shift, n4);
}